// WeightedChamferDistance_31799937859588
// MI455X (gfx1250) — compile-verified
//
#include <hip/hip_runtime.h>

// Weighted Chamfer distance, B=4, N=M=8192, D=3.
// d2 tile = ONE v_wmma_f32_16x16x4_f32 via embedding:
//   A row  (source n): [sx, sy, sz, |s|^2]
//   B col  (target m): [-2tx, -2ty, -2tz, 1]   (precomputed, packed float4)
//   C      (accum)   : |t_m|^2 broadcast down each column
// Forward min: per-lane running min + shfl reduce (once per wave, outside loop).
// Backward min: LDS ds_min_u32 on sortable-uint encoding (merges row-halves via
// same-address atomics), flushed once per block. Inner loop software-pipelined
// so each tile's b128 load is covered by the previous tile's WMMA+min work.

typedef __attribute__((ext_vector_type(2))) float v2f;
typedef __attribute__((ext_vector_type(8))) float v8f;

#define BB 4
#define NN 8192
#define MM 8192
#define SLICES 4
#define MT_PER_SLICE (MM / 16 / SLICES)   // 128 m-tiles per slice
#define COLS_PER_SLICE (MM / SLICES)      // 2048 columns per slice

// Monotone float -> uint encoding (total order matches float order).
__device__ __forceinline__ unsigned encodeF(float f) {
    int bi = __float_as_int(f);
    return (bi >= 0) ? (unsigned(bi) ^ 0x80000000u) : ~unsigned(bi);
}
__device__ __forceinline__ float decodeF(unsigned u) {
    int bi = (u & 0x80000000u) ? int(u ^ 0x80000000u) : int(~u);
    return __int_as_float(bi);
}

// One 16x16 distance tile: WMMA + fwd/bwd mins + LDS atomic-min.
__device__ __forceinline__ void tile_step(const float4& p, const v2f& A, int half,
                                          float fm[8], unsigned* sminp) {
    v2f Bm;
    Bm[0] = half ? p.z : p.x;
    Bm[1] = half ? 1.0f : p.y;
    v8f C;
#pragma unroll
    for (int r = 0; r < 8; ++r) C[r] = p.w;

    // D[n][m] = |s|^2 + |t|^2 - 2 s.t
    v8f D = __builtin_amdgcn_wmma_f32_16x16x4_f32(
        false, A, false, Bm, (short)0, C, false, false);

    float bcol = fminf(fminf(fminf(D[0], D[1]), fminf(D[2], D[3])),
                       fminf(fminf(D[4], D[5]), fminf(D[6], D[7])));
#pragma unroll
    for (int r = 0; r < 8; ++r) fm[r] = fminf(fm[r], D[r]);

    // Lanes l and l+16 hit the same word -> DS atomic merges the row halves.
    atomicMin(sminp, encodeF(bcol));
}

// ---- Kernel 1: pack target operands + init workspace --------------------------
__global__ void __launch_bounds__(256) prep_kernel(const float* __restrict__ tgt,
                                                   float4* __restrict__ PT,
                                                   unsigned* __restrict__ bwdmin,
                                                   float* __restrict__ acc) {
    int i = blockIdx.x * 256 + threadIdx.x;
    if (i < BB * MM) {
        const float* tp = tgt + (size_t)i * 3;
        float x = tp[0], y = tp[1], z = tp[2];
        float t2 = fmaf(x, x, fmaf(y, y, z * z));
        PT[i] = make_float4(-2.0f * x, -2.0f * y, -2.0f * z, t2);
        bwdmin[i] = 0xFFFFFFFFu;     // encoded "+NaN" > every real value
    }
    if (i == 0) *acc = 0.0f;
}

// ---- Kernel 2: main WMMA sweep ------------------------------------------------
// grid = (NN/128, SLICES, BB); block = 256 (8 waves, one 16-row n-tile each)
__global__ void __launch_bounds__(256) chamfer_wmma_kernel(
    const float* __restrict__ src,     // [B, N, 3]
    const float4* __restrict__ PT,     // [B, M] packed (-2t, |t|^2)
    unsigned* __restrict__ bwdmin,     // [B, M] encoded global mins
    float* __restrict__ fwdpart)       // [B, SLICES, N] per-slice row mins
{
    __shared__ unsigned smin[COLS_PER_SLICE];   // 8 KB

    const int lane  = threadIdx.x & 31;
    const int wave  = threadIdx.x >> 5;
    const int b     = blockIdx.z;
    const int slice = blockIdx.y;
    const int nbase = (blockIdx.x * 8 + wave) * 16;
    const int half  = lane >> 4;    // 0: K=0,1 / rows r ; 1: K=2,3 / rows r+8
    const int l15   = lane & 15;

    for (int i = threadIdx.x; i < COLS_PER_SLICE; i += 256) smin[i] = 0xFFFFFFFFu;
    __syncthreads();

    // A matrix (16x4 f32): lanes 0-15 -> (K0,K1)=(sx,sy); lanes 16-31 -> (K2,K3)=(sz,|s|^2)
    const float* sp = src + ((size_t)b * NN + (size_t)(nbase + l15)) * 3;
    float sx = sp[0], sy = sp[1], sz = sp[2];
    float s2 = fmaf(sx, sx, fmaf(sy, sy, sz * sz));
    v2f A;
    A[0] = half ? sz : sx;
    A[1] = half ? s2 : sy;

    float fm[8];
#pragma unroll
    for (int r = 0; r < 8; ++r) fm[r] = __builtin_inff();

    const float4* pt = PT + (size_t)b * MM + (size_t)slice * COLS_PER_SLICE;

    // Software-pipelined sweep: issue tile k+1's load, then compute tile k.
    float4 p = pt[l15];
#pragma unroll 4
    for (int mt = 0; mt < MT_PER_SLICE - 1; ++mt) {
        float4 pn = pt[((mt + 1) << 4) + l15];        // prefetch next tile's operand
        tile_step(p, A, half, fm, &smin[(mt << 4) + l15]);
        p = pn;
    }
    tile_step(p, A, half, fm, &smin[((MT_PER_SLICE - 1) << 4) + l15]);

    // Forward: reduce across the 16 column-lanes (stays inside each half-wave).
#pragma unroll
    for (int r = 0; r < 8; ++r) {
        float v = fm[r];
        v = fminf(v, __shfl_xor(v, 1));
        v = fminf(v, __shfl_xor(v, 2));
        v = fminf(v, __shfl_xor(v, 4));
        v = fminf(v, __shfl_xor(v, 8));
        fm[r] = v;
    }
    if (l15 == 0) {
        // lane 0 -> rows nbase..+7 ; lane 16 -> rows nbase+8..+15 (race-free store)
        float* fp = fwdpart + ((size_t)(b * SLICES + slice)) * NN + nbase + half * 8;
#pragma unroll
        for (int r = 0; r < 8; ++r) fp[r] = fm[r];
    }

    // Flush block-local backward mins to global (once per column per block).
    __syncthreads();
    unsigned* gb = bwdmin + (size_t)b * MM + (size_t)slice * COLS_PER_SLICE;
    for (int i = threadIdx.x; i < COLS_PER_SLICE; i += 256)
        atomicMin(&gb[i], smin[i]);
}

// ---- Kernel 3: combine forward slice-mins, weight, accumulate -----------------
__global__ void __launch_bounds__(256) fwd_reduce_kernel(const float* __restrict__ fwdpart,
                                                         const float* __restrict__ wsrc,
                                                         float* __restrict__ acc) {
    __shared__ float sdata[256];
    int i = blockIdx.x * 256 + threadIdx.x;    // over B*N
    int b = i / NN, n = i % NN;
    float v = __builtin_inff();
#pragma unroll
    for (int s = 0; s < SLICES; ++s)
        v = fminf(v, fwdpart[((size_t)(b * SLICES + s)) * NN + n]);
    sdata[threadIdx.x] = v * wsrc[i];
    __syncthreads();
    for (int off = 128; off > 0; off >>= 1) {
        if ((int)threadIdx.x < off) sdata[threadIdx.x] += sdata[threadIdx.x + off];
        __syncthreads();
    }
    if (threadIdx.x == 0) atomicAdd(acc, sdata[0]);
}

// ---- Kernel 4: sum backward mins + finalize scalar ----------------------------
__global__ void __launch_bounds__(256) finalize_kernel(const unsigned* __restrict__ bwdmin,
                                                       const float* __restrict__ acc,
                                                       float* __restrict__ out) {
    __shared__ float sdata[256];
    float s = 0.0f;
    for (int i = threadIdx.x; i < BB * MM; i += 256) s += decodeF(bwdmin[i]);
    sdata[threadIdx.x] = s;
    __syncthreads();
    for (int off = 128; off > 0; off >>= 1) {
        if ((int)threadIdx.x < off) sdata[threadIdx.x] += sdata[threadIdx.x + off];
        __syncthreads();
    }
    if (threadIdx.x == 0) out[0] = (sdata[0] + *acc) * (1.0f / (float)BB);
}

extern "C" void kernel_launch(void* const* d_in, const int* in_sizes, int n_in,
                              void* d_out, int out_size, void* d_ws, size_t ws_size,
                              hipStream_t stream) {
    (void)in_sizes; (void)n_in; (void)out_size; (void)ws_size;
    const float* src  = (const float*)d_in[0];   // [B,N,3]
    const float* tgt  = (const float*)d_in[1];   // [B,M,3]
    const float* wsrc = (const float*)d_in[2];   // [B,N]

    char* ws = (char*)d_ws;
    float4*   PT      = (float4*)ws;                                  ws += (size_t)BB * MM * sizeof(float4);
    unsigned* bwdmin  = (unsigned*)ws;                                ws += (size_t)BB * MM * sizeof(unsigned);
    float*    fwdpart = (float*)ws;                                   ws += (size_t)BB * SLICES * NN * sizeof(float);
    float*    acc     = (float*)ws;
    float*    out     = (float*)d_out;

    prep_kernel<<<(BB * MM + 255) / 256, 256, 0, stream>>>(tgt, PT, bwdmin, acc);

    dim3 grid(NN / (16 * 8), SLICES, BB);   // 1024 blocks, 8192 waves
    chamfer_wmma_kernel<<<grid, 256, 0, stream>>>(src, PT, bwdmin, fwdpart);

    fwd_reduce_kernel<<<(BB * NN) / 256, 256, 0, stream>>>(fwdpart, wsrc, acc);
    finalize_kernel<<<1, 256, 0, stream>>>(bwdmin, acc, out);
}